// TransformNRF_6073083756912
// MI455X (gfx1250) — compile-verified
//
#include <hip/hip_runtime.h>
#include <math.h>

// out[b, i, p] = NRF[b, p] * 0.5  if atom i in pair p (tril order), else 0.
// B=1024, N_ATOMS=64, NC2=2016. Output 528 MB f32 -> pure store-BW bound
// (~22.7 us floor @ 23.3 TB/s). Tile computation done with
// V_WMMA_F32_16X16X4_F32 (exact f32 math: mask is 0/1, exactly one nonzero
// product per output element), 4 accumulating K=4 WMMAs per 16x16 tile.

#define N_ATOMS  64
#define NC2      2016
#define BATCH    1024
#define P_TILES  126     // 2016 / 16
#define I_TILES  4       // 64 / 16
#define B_CHUNK  64
#define B_CHUNKS 16      // 1024 / 64
// total wave tasks = 126 * 4 * 16 = 8064 -> 1008 blocks x 8 waves

typedef __attribute__((ext_vector_type(2))) float v2f;
typedef __attribute__((ext_vector_type(8))) float v8f;

// p -> i2 (row of tril_indices(k=-1)); fp32 sqrt seed + exact integer fixup
__device__ __forceinline__ int decode_i2(int p) {
  int i2 = (int)((1.0f + sqrtf(8.0f * (float)p + 1.0f)) * 0.5f);
  while (i2 * (i2 - 1) / 2 > p) --i2;
  while ((i2 + 1) * i2 / 2 <= p) ++i2;
  return i2;
}

__device__ __forceinline__ float mask_val(int i, int p) {
  const int i2 = decode_i2(p);
  const int j2 = p - i2 * (i2 - 1) / 2;
  return (i == i2 || i == j2) ? 1.0f : 0.0f;
}

__global__ __launch_bounds__(256) void eij_nrf_kernel(
    const float* __restrict__ nrf, float* __restrict__ out) {
  const int lane = threadIdx.x & 31;
  const int wid  = blockIdx.x * (blockDim.x >> 5) + (threadIdx.x >> 5);

  // wid -> (p_tile, i_tile, b_chunk); exactly 8064 waves, no tail
  const int pt = wid % P_TILES;
  const int t  = wid / P_TILES;
  const int it = t % I_TILES;
  const int bc = t / I_TILES;

  const int p0 = pt * 16;
  const int i0 = it * 16;
  const int b0 = bc * B_CHUNK;

  const int n  = lane & 15;   // N index (p within tile); also A's M index
  const int hi = lane >> 4;   // lane-half: selects K pairs {0,1} vs {2,3}

  // Loop-invariant per-lane WMMA operand state.
  // A (16x4 f32, ISA layout): vgpr0 holds K = 2*hi, vgpr1 holds K = 2*hi+1,
  //   M = lane&15.  A[m,k] = mask[i0+m, p0+4q+k].
  // B (4x16 f32, mirror layout assumed): vgpr0 K = 2*hi, vgpr1 K = 2*hi+1,
  //   N = lane&15.  B_q[k,n] = 0.5*val_raw[p0+n] iff n == 4q+k  (0.5 folded in).
  float am0[4], am1[4], bs0[4], bs1[4];
#pragma unroll
  for (int q = 0; q < 4; ++q) {
    const int k0 = 4 * q + 2 * hi;
    const int k1 = k0 + 1;
    am0[q] = mask_val(i0 + n, p0 + k0);
    am1[q] = mask_val(i0 + n, p0 + k1);
    bs0[q] = (n == k0) ? 0.5f : 0.0f;
    bs1[q] = (n == k1) ? 0.5f : 0.0f;
  }

  const float* nrf_ptr = nrf + (long)b0 * NC2 + p0 + n;
  // D layout: VGPR r -> row M = r + 8*hi, col N = lane&15
  float* out_ptr = out + ((long)(b0 * N_ATOMS + i0 + 8 * hi) * NC2 + p0 + n);

  for (int b = 0; b < B_CHUNK; ++b) {
    const float val = nrf_ptr[0];   // L2-resident, reused across i-tiles
    v8f c = {};
#pragma unroll
    for (int q = 0; q < 4; ++q) {
      v2f a;  a.x = am0[q];       a.y = am1[q];
      v2f bb; bb.x = bs0[q] * val; bb.y = bs1[q] * val;
      // (neg_a, A, neg_b, B, c_mod, C, reuse_a, reuse_b)
      c = __builtin_amdgcn_wmma_f32_16x16x4_f32(
          false, a, false, bb, (short)0, c, false, false);
    }
#pragma unroll
    for (int r = 0; r < 8; ++r) {
      // 528 MB stream >> 192 MB L2: non-temporal stores
      __builtin_nontemporal_store(c[r], out_ptr + (long)r * NC2);
    }
    nrf_ptr += NC2;
    out_ptr += (long)N_ATOMS * NC2;
  }
}

extern "C" void kernel_launch(void* const* d_in, const int* in_sizes, int n_in,
                              void* d_out, int out_size, void* d_ws, size_t ws_size,
                              hipStream_t stream) {
  const float* nrf = (const float*)d_in[0];
  float* out = (float*)d_out;
  (void)in_sizes; (void)n_in; (void)out_size; (void)d_ws; (void)ws_size;
  // 8064 waves = 1008 blocks * 8 wave32s; full output coverage, no fill pass
  eij_nrf_kernel<<<dim3(1008), dim3(256), 0, stream>>>(nrf, out);
}